// SpatialProductLayer_49323404427648
// MI455X (gfx1250) — compile-verified
//
#include <hip/hip_runtime.h>
#include <stdint.h>

// Problem constants (from the reference module config)
#define CIN   64
#define H     64
#define W     64
#define OC    256
#define OH    65
#define OW    65
#define KTAPS 4        // 2x2 kernel

// Tiling
#define TY    2        // output rows per block
#define ROWS  3        // padded input rows staged = TY+1
#define PITCH 72       // floats per staged row; padded col pc lives at slot pc+3
                       // -> global col 0 at slot 4 => byte offset 16-aligned for B128

// ---------------------------------------------------------------------------
// Kernel 1: recover the one-hot channel selection indices from the 0/1 weight.
// weight layout (O, I, KH, KW); idx[o*4 + (kh*2+kw)] = selected input channel.
// ---------------------------------------------------------------------------
__global__ __launch_bounds__(256)
void build_idx_kernel(const float* __restrict__ w, int* __restrict__ idx) {
  int o = threadIdx.x;
  if (o >= OC) return;
  int sel0 = 0, sel1 = 0, sel2 = 0, sel3 = 0;
  for (int c = 0; c < CIN; ++c) {
    const float* p = w + ((size_t)o * CIN + c) * KTAPS;
    if (p[0] > 0.5f) sel0 = c;
    if (p[1] > 0.5f) sel1 = c;
    if (p[2] > 0.5f) sel2 = c;
    if (p[3] > 0.5f) sel3 = c;
  }
  idx[o * 4 + 0] = sel0;
  idx[o * 4 + 1] = sel1;
  idx[o * 4 + 2] = sel2;
  idx[o * 4 + 3] = sel3;
}

// ---------------------------------------------------------------------------
// Kernel 2: gather-sum. Stage input tile (64ch x 3 padded rows) into LDS via
// CDNA5 async global->LDS loads, then compute 256 output channels from LDS.
// ---------------------------------------------------------------------------
__global__ __launch_bounds__(256)
void spatial_product_kernel(const float* __restrict__ x,
                            const int* __restrict__ idx,
                            float* __restrict__ out) {
  __shared__ float tile[CIN * ROWS * PITCH];   // 55296 B
  __shared__ int4  sidx[OC];                   //  4096 B

  const int tid = threadIdx.x;
  const int b   = blockIdx.y;
  const int y0  = blockIdx.x * TY;             // first output row of this tile
  const int nry = (OH - y0) < TY ? (OH - y0) : TY;  // valid output rows (2 or 1)

  // ---- Phase 1: zero-fill tile (implements the 'full' padding) + idx table
  {
    float4  z4 = make_float4(0.f, 0.f, 0.f, 0.f);
    float4* t4 = (float4*)tile;
    #pragma unroll 2
    for (int i = tid; i < (CIN * ROWS * PITCH) / 4; i += 256) t4[i] = z4;
    sidx[tid] = ((const int4*)idx)[tid];       // blockDim.x == 256 == OC
  }
  __syncthreads();

  // ---- Phase 2: async-stage valid interior rows: global -> LDS (B128/lane)
  const float* xb    = x + (size_t)b * CIN * H * W;       // uniform per block
  unsigned     lbase = (unsigned)(uintptr_t)(void*)tile;  // low 32b = LDS offset
  for (int ry = 0; ry <= nry; ++ry) {
    int gr = y0 + ry - 1;                      // padded row -> global row
    if (gr < 0 || gr > H - 1) continue;        // stays zero (pad row)
    // 64 channels x 16 groups of 4 floats = 1024 lane-transfers, 4 per thread
    for (int m = tid; m < CIN * 16; m += 256) {
      int ch = m >> 4;
      int cg = m & 15;
      unsigned ldsoff = lbase +
          (unsigned)(((ch * ROWS + ry) * PITCH + 4 + cg * 4) * 4); // 16B aligned
      unsigned goff = (unsigned)(((ch * H + gr) * W + cg * 4) * 4);
      asm volatile("global_load_async_to_lds_b128 %0, %1, %2"
                   :: "v"(ldsoff), "v"(goff), "s"(xb)
                   : "memory");
    }
  }
  asm volatile("s_wait_asynccnt 0" ::: "memory");
  __syncthreads();

  // ---- Phase 3: gather 4 taps from LDS, sum, streaming (NT) store
  for (int n = tid; n < OC * TY * OW; n += 256) {
    int o  = n / (TY * OW);                    // constant divisor -> mul/shift
    int r  = n - o * (TY * OW);
    int ry = r / OW;
    int xx = r - ry * OW;
    if (ry >= nry) continue;                   // last tile has 1 row
    int4 cs = sidx[o];                         // (kh0kw0, kh0kw1, kh1kw0, kh1kw1)
    float v = tile[(cs.x * ROWS + ry    ) * PITCH + xx + 3]
            + tile[(cs.y * ROWS + ry    ) * PITCH + xx + 4]
            + tile[(cs.z * ROWS + ry + 1) * PITCH + xx + 3]
            + tile[(cs.w * ROWS + ry + 1) * PITCH + xx + 4];
    size_t oi = (((size_t)b * OC + o) * OH + (size_t)(y0 + ry)) * OW + xx;
    __builtin_nontemporal_store(v, &out[oi]);  // keep x L2-resident
  }
}

// ---------------------------------------------------------------------------
extern "C" void kernel_launch(void* const* d_in, const int* in_sizes, int n_in,
                              void* d_out, int out_size, void* d_ws, size_t ws_size,
                              hipStream_t stream) {
  const float* x   = (const float*)d_in[0];   // (64, 64, 64, 64) f32
  const float* w   = (const float*)d_in[1];   // (256, 64, 2, 2) f32 one-hot
  float*       out = (float*)d_out;           // (64, 256, 65, 65) f32
  int*         idx = (int*)d_ws;              // 256*4 ints of scratch

  build_idx_kernel<<<1, 256, 0, stream>>>(w, idx);

  dim3 grid((OH + TY - 1) / TY, 64);          // 33 row-tiles x 64 batches
  spatial_product_kernel<<<grid, 256, 0, stream>>>(x, idx, out);
}